// SWDLoss_42537356100179
// MI455X (gfx1250) — compile-verified
//
#include <hip/hip_runtime.h>
#include <hip/hip_bf16.h>

typedef float v2f __attribute__((ext_vector_type(2)));
typedef float v4f __attribute__((ext_vector_type(4)));
typedef float v8f __attribute__((ext_vector_type(8)));
typedef unsigned int v4u __attribute__((ext_vector_type(4)));
typedef int v4i __attribute__((ext_vector_type(4)));
typedef int v8i __attribute__((ext_vector_type(8)));

#define B_N 16384
#define F_N 512
#define NP  128

#if defined(__gfx1250__) && __has_builtin(__builtin_amdgcn_tensor_load_to_lds) && __has_builtin(__builtin_amdgcn_s_wait_tensorcnt)
#define USE_TDM 1
#else
#define USE_TDM 0
#endif

#if USE_TDM
// ---------------------------------------------------------------------------
// Issue a TDM DMA of one contiguous row of 16384 f32 (64 KB) from global
// memory into LDS offset 0. D# layout per CDNA5 ISA ch. 8 (CDNA5 ISA):
//  group0: count=1 | lds_addr=0 | global_addr[56:0] in bits 120:64 | type=2
//  group1: data_size=2 (4B), tensor_dim0=16384, tensor_dim1=1,
//          tile_dim0=16384, tile_dim1=1, tensor_dim0_stride=16384
// This toolchain's builtin is the 6-arg form:
//  (v4u g0, v8i g1, v4i g2, v4i g3, v8i g4, i32 cpol)
// ---------------------------------------------------------------------------
__device__ __forceinline__ void tdm_load_row_to_lds(const float* gsrc) {
    unsigned long long ga = (unsigned long long)gsrc;
    v4u g0 = { 1u,                                   // count=1, is_restore=0, gather=0
               0u,                                   // lds_addr = 0 (dynamic LDS base)
               (unsigned)(ga & 0xffffffffull),       // global_addr[31:0]
               (unsigned)((ga >> 32) & 0x01ffffffull) | 0x80000000u }; // addr[56:32] | type=2
    v8i g1 = { 0x00020000,   // workgroup_mask=0, data_size=2 (4 bytes)
               0x40000000,   // tensor_dim0[15:0]=0x4000 in bits 31:16 (atomic_barrier=0)
               0x00010000,   // tensor_dim0[31:16]=0, tensor_dim1[15:0]=1
               0x40000000,   // tensor_dim1[31:16]=0, tile_dim0=0x4000
               0x00000001,   // tile_dim1=1, tile_dim2=0
               16384,        // tensor_dim0_stride[31:0]
               0, 0 };       // stride0 hi / tensor_dim1_stride = 0
    v4i gz4 = { 0, 0, 0, 0 };
    v8i gz8 = { 0, 0, 0, 0, 0, 0, 0, 0 };
    __builtin_amdgcn_tensor_load_to_lds(g0, g1, gz4, gz4, gz8, 0);
}
#endif

// ---------------------------------------------------------------------------
// K1: normalize projection columns. P[f*NP+c] = projs[f*NP+c] / ||col c||
// ---------------------------------------------------------------------------
__global__ __launch_bounds__(128) void k_norm_projs(const float* __restrict__ projs,
                                                    float* __restrict__ P) {
    int c = threadIdx.x;                     // column 0..127
    float s = 0.f;
    for (int f = 0; f < F_N; ++f) { float v = projs[f * NP + c]; s += v * v; }
    float r = 1.0f / sqrtf(s);
    for (int f = 0; f < F_N; ++f) P[f * NP + c] = projs[f * NP + c] * r;
}

// ---------------------------------------------------------------------------
// K2: src_pT/tgt_pT [NP,B] = (X[B,F] @ P[F,NP])^T via V_WMMA_F32_16X16X4_F32.
// One wave computes one 16x16 tile for BOTH src and tgt (shared B fragment).
// One-deep software pipeline: next-k fragments fetched before current WMMAs.
// ---------------------------------------------------------------------------
__global__ __launch_bounds__(256) void k_proj_gemm(const float* __restrict__ src,
                                                   const float* __restrict__ tgt,
                                                   const float* __restrict__ P,
                                                   float* __restrict__ srcT,
                                                   float* __restrict__ tgtT) {
    int wave = (int)((blockIdx.x * blockDim.x + threadIdx.x) >> 5);
    int lane = threadIdx.x & 31;
    int tile_p = wave & 7;                   // 8 tiles along NP
    int tile_b = wave >> 3;                  // 1024 tiles along B
    int kh  = lane >> 4;                     // half-wave: 0/1
    int l16 = lane & 15;
    int brow = tile_b * 16 + l16;            // A-matrix row (M)
    int pcol = tile_p * 16 + l16;            // B-matrix col (N)

    v8f cs = {}; v8f ct = {};
    const float* srow = src + (size_t)brow * F_N;
    const float* trow = tgt + (size_t)brow * F_N;

    // prologue: fragments for k = 0
    int k0 = 2 * kh;                         // A: lanes0-15 K=0,1 ; lanes16-31 K=2,3
    v2f a_s = *(const v2f*)(srow + k0);
    v2f a_t = *(const v2f*)(trow + k0);
    v2f bf;
    bf.x = P[(size_t)k0 * NP + pcol];
    bf.y = P[(size_t)(k0 + 1) * NP + pcol];

    for (int k = 0; k < F_N; k += 4) {
        int kn = k + 4;
        int kp = (kn < F_N) ? (kn + 2 * kh) : (2 * kh);   // clamped prefetch index
        v2f na_s = *(const v2f*)(srow + kp);
        v2f na_t = *(const v2f*)(trow + kp);
        v2f nbf;
        nbf.x = P[(size_t)kp * NP + pcol];
        nbf.y = P[(size_t)(kp + 1) * NP + pcol];

        cs = __builtin_amdgcn_wmma_f32_16x16x4_f32(false, a_s, false, bf, (short)0, cs, false, false);
        ct = __builtin_amdgcn_wmma_f32_16x16x4_f32(false, a_t, false, bf, (short)0, ct, false, false);

        a_s = na_s; a_t = na_t; bf = nbf;
    }
    // C layout: VGPR r -> M = r + 8*kh, N = l16. Transposed store: contiguous 8 floats.
    size_t off = (size_t)pcol * B_N + (size_t)tile_b * 16 + 8 * kh;
    union { v8f v; v4f q[2]; } us, ut;
    us.v = cs; ut.v = ct;
    *(v4f*)(srcT + off)     = us.q[0];
    *(v4f*)(srcT + off + 4) = us.q[1];
    *(v4f*)(tgtT + off)     = ut.q[0];
    *(v4f*)(tgtT + off + 4) = ut.q[1];
}

// ---------------------------------------------------------------------------
// K3a: in-place ascending bitonic sort of one projection slice (16384 floats)
// entirely in LDS. Slice is staged into LDS by the Tensor Data Mover.
// ---------------------------------------------------------------------------
__global__ __launch_bounds__(1024) void k_sort_vals(float* __restrict__ data) {
    extern __shared__ float skey[];
    const int N = B_N, T = 1024;
    float* g = data + (size_t)blockIdx.x * N;
#if USE_TDM
    if (threadIdx.x < 32) tdm_load_row_to_lds(g);     // one DMA per workgroup
    __builtin_amdgcn_s_wait_tensorcnt(0);             // all waves drain TENSORcnt
    __syncthreads();
#else
    for (int i = threadIdx.x; i < N; i += T) skey[i] = g[i];
    __syncthreads();
#endif
    for (int k = 2; k <= N; k <<= 1) {
        for (int j = k >> 1; j > 0; j >>= 1) {
            for (int i = threadIdx.x; i < N; i += T) {
                int ixj = i ^ j;
                if (ixj > i) {
                    float a = skey[i], b = skey[ixj];
                    bool up = (i & k) == 0;
                    bool sw = up ? (a > b) : (a < b);
                    if (sw) { skey[i] = b; skey[ixj] = a; }
                }
            }
            __syncthreads();
        }
    }
    for (int i = threadIdx.x; i < N; i += T) g[i] = skey[i];
}

// ---------------------------------------------------------------------------
// K3b: key+index bitonic sort of src slice (128 KB LDS), then scatter
//   delta[orig_idx[j]] = tgt_sorted[j] - src_sorted[j]   (in place over src)
// plus fixed-order block reduction of sum(delta^2) -> partials[p].
// Keys staged via TDM; indices synthesized in LDS.
// ---------------------------------------------------------------------------
__global__ __launch_bounds__(1024) void k_sort_src_delta(float* __restrict__ srcdelta,
                                                         const float* __restrict__ tgtsorted,
                                                         float* __restrict__ partials) {
    extern __shared__ float smem2[];
    float* skey = smem2;
    int*   sidx = (int*)(smem2 + B_N);
    const int N = B_N, T = 1024;
    float* g        = srcdelta  + (size_t)blockIdx.x * N;
    const float* ts = tgtsorted + (size_t)blockIdx.x * N;

#if USE_TDM
    if (threadIdx.x < 32) tdm_load_row_to_lds(g);     // keys -> first 64 KB of LDS
    for (int i = threadIdx.x; i < N; i += T) sidx[i] = i;
    __builtin_amdgcn_s_wait_tensorcnt(0);
    __syncthreads();
#else
    for (int i = threadIdx.x; i < N; i += T) { skey[i] = g[i]; sidx[i] = i; }
    __syncthreads();
#endif
    for (int k = 2; k <= N; k <<= 1) {
        for (int j = k >> 1; j > 0; j >>= 1) {
            for (int i = threadIdx.x; i < N; i += T) {
                int ixj = i ^ j;
                if (ixj > i) {
                    float a = skey[i], b = skey[ixj];
                    int ia = sidx[i], ib = sidx[ixj];
                    bool gt = (a > b) || (a == b && ia > ib);
                    bool up = (i & k) == 0;
                    if (gt == up) { skey[i] = b; skey[ixj] = a; sidx[i] = ib; sidx[ixj] = ia; }
                }
            }
            __syncthreads();
        }
    }
    float acc = 0.f;
    for (int i = threadIdx.x; i < N; i += T) {
        float d = ts[i] - skey[i];           // sorted tgt minus sorted src
        g[sidx[i]] = d;                      // scatter back to original position
        acc += d * d;
    }
    __syncthreads();
    skey[threadIdx.x] = acc;
    __syncthreads();
    for (int s = T / 2; s > 0; s >>= 1) {
        if ((int)threadIdx.x < s) skey[threadIdx.x] += skey[threadIdx.x + s];
        __syncthreads();
    }
    if (threadIdx.x == 0) partials[blockIdx.x] = skey[0];
}

// ---------------------------------------------------------------------------
// K4: deterministic final reduction -> dist at d_out[0]
// ---------------------------------------------------------------------------
__global__ __launch_bounds__(128) void k_dist(const float* __restrict__ partials,
                                              float* __restrict__ out) {
    __shared__ float s[128];
    s[threadIdx.x] = partials[threadIdx.x];
    __syncthreads();
    for (int st = 64; st > 0; st >>= 1) {
        if ((int)threadIdx.x < st) s[threadIdx.x] += s[threadIdx.x + st];
        __syncthreads();
    }
    if (threadIdx.x == 0) out[0] = s[0] / (float)((size_t)NP * B_N);
}

// ---------------------------------------------------------------------------
// K5: movement[B,F] = delta^T[B,NP] @ P^T[NP,F] via V_WMMA_F32_16X16X4_F32,
// with one-deep software pipeline.
// ---------------------------------------------------------------------------
__global__ __launch_bounds__(256) void k_move_gemm(const float* __restrict__ delta,
                                                   const float* __restrict__ P,
                                                   float* __restrict__ mov) {
    int wave = (int)((blockIdx.x * blockDim.x + threadIdx.x) >> 5);
    int lane = threadIdx.x & 31;
    int tile_f = wave & 31;                  // 32 tiles along F
    int tile_b = wave >> 5;                  // 1024 tiles along B
    int kh  = lane >> 4;
    int l16 = lane & 15;
    int brow = tile_b * 16 + l16;
    int fcol = tile_f * 16 + l16;

    v8f c = {};
    int k0 = 2 * kh;
    v2f a;
    a.x = delta[(size_t)k0 * B_N + brow];
    a.y = delta[(size_t)(k0 + 1) * B_N + brow];
    v2f bf = *(const v2f*)(P + (size_t)fcol * NP + k0);

    for (int k = 0; k < NP; k += 4) {
        int kn = k + 4;
        int kp = (kn < NP) ? (kn + 2 * kh) : (2 * kh);
        v2f na;
        na.x = delta[(size_t)kp * B_N + brow];
        na.y = delta[(size_t)(kp + 1) * B_N + brow];
        v2f nbf = *(const v2f*)(P + (size_t)fcol * NP + kp);

        c = __builtin_amdgcn_wmma_f32_16x16x4_f32(false, a, false, bf, (short)0, c, false, false);

        a = na; bf = nbf;
    }
    int row0 = tile_b * 16 + 8 * kh;
    union { v8f v; float f[8]; } u; u.v = c;
    #pragma unroll
    for (int r = 0; r < 8; ++r)
        mov[(size_t)(row0 + r) * F_N + fcol] = u.f[r];
}

// ---------------------------------------------------------------------------
// K6: per-row L2 normalize movement -> d_out+1 (scalar stores: d_out+1 is
// only 4-byte aligned). 512 threads = one element each; LDS tree reduce.
// ---------------------------------------------------------------------------
__global__ __launch_bounds__(512) void k_rownorm(const float* __restrict__ mov,
                                                 float* __restrict__ out) {
    __shared__ float s[512];
    int b = blockIdx.x;
    float v = mov[(size_t)b * F_N + threadIdx.x];
    s[threadIdx.x] = v * v;
    __syncthreads();
    for (int st = 256; st > 0; st >>= 1) {
        if ((int)threadIdx.x < st) s[threadIdx.x] += s[threadIdx.x + st];
        __syncthreads();
    }
    float r = 1.0f / sqrtf(s[0]);
    out[(size_t)b * F_N + threadIdx.x] = v * r;
}

// ---------------------------------------------------------------------------
extern "C" void kernel_launch(void* const* d_in, const int* in_sizes, int n_in,
                              void* d_out, int out_size, void* d_ws, size_t ws_size,
                              hipStream_t stream) {
    const float* src   = (const float*)d_in[0];   // [B,F]
    const float* tgt   = (const float*)d_in[1];   // [B,F]
    const float* projs = (const float*)d_in[2];   // [F,NP]
    float* out = (float*)d_out;                   // [0]=dist, [1..]=movement_norm

    float* ws       = (float*)d_ws;
    float* P        = ws;                                   // F*NP      = 64K floats
    float* srcT     = P    + (size_t)F_N * NP;              // NP*B      = 2M floats (becomes delta)
    float* tgtT     = srcT + (size_t)NP * B_N;              // NP*B      = 2M floats (becomes tgt_sorted)
    float* mov      = tgtT + (size_t)NP * B_N;              // B*F       = 8M floats
    float* partials = mov  + (size_t)B_N * F_N;             // NP floats

    k_norm_projs<<<1, 128, 0, stream>>>(projs, P);

    // (B/16)*(NP/16) = 8192 waves, 8 waves/block
    k_proj_gemm<<<1024, 256, 0, stream>>>(src, tgt, P, srcT, tgtT);

    k_sort_vals<<<NP, 1024, B_N * sizeof(float), stream>>>(tgtT);
    k_sort_src_delta<<<NP, 1024, B_N * (sizeof(float) + sizeof(int)), stream>>>(srcT, tgtT, partials);

    k_dist<<<1, 128, 0, stream>>>(partials, out);

    // (B/16)*(F/16) = 32768 waves, 8 waves/block
    k_move_gemm<<<4096, 256, 0, stream>>>(srcT, P, mov);

    k_rownorm<<<B_N, 512, 0, stream>>>(mov, out + 1);
}